// GATLayer_31722628448598
// MI455X (gfx1250) — compile-verified
//
#include <hip/hip_runtime.h>
#include <math.h>

// ---------------------------------------------------------------------------
// GAT layer for MI455X (gfx1250).  fp32 throughout; matrix ops use
// V_WMMA_F32_16X16X4_F32 (wave32, 16x16 f32 tile per wave, K-step 4).
// Memory-bound on the 512MB edge_attr stream -> fuse score+mask+softmax.
// P@h GEMM is LDS-blocked: 8-wave block owns a 128x128 output stripe,
// h staged in 32KB LDS chunks, each A-fragment feeds 8 WMMAs.
// ---------------------------------------------------------------------------

#define B_    8
#define N_    1024
#define IND_  128
#define OUTD_ 128
#define ED_   16
#define ALPHA_ 0.2f
#define NEG_INF_ (-9.0e15f)
#define KC_   64          // K-chunk of h staged in LDS (64*128*4B = 32KB)

typedef __attribute__((ext_vector_type(2))) float v2f;
typedef __attribute__((ext_vector_type(8))) float v8f;

// ---------------------------------------------------------------------------
// Fragment layouts per CDNA5 ISA 7.12.2 (V_WMMA_F32_16X16X4_F32):
//   A (16x4 f32, 2 VGPRs): lanes 0-15 hold M=lane, {K=k+0,k+1}; lanes 16-31
//     hold M=lane-16, {K=k+2,k+3}  -> one contiguous float2 per lane.
//   B (4x16 f32, 2 VGPRs): one K-row striped across 16 lanes per VGPR-half:
//     v0: rows k+0 (lanes 0-15) / k+1 (lanes 16-31); v1: rows k+2 / k+3.
//   C/D (8 VGPRs): VGPR j -> row m0+j (lanes 0-15), row m0+j+8 (lanes 16-31),
//     col n0 + (lane&15).
// ---------------------------------------------------------------------------
__device__ __forceinline__
v8f wmma_f32_tile(const float* __restrict__ A, int lda,
                  const float* __restrict__ B, int ldb,
                  int K, int m0, int n0, int lane) {
  const int half = lane >> 4;
  const int l16  = lane & 15;
  v8f acc = {0.f, 0.f, 0.f, 0.f, 0.f, 0.f, 0.f, 0.f};
  const float* ap = A + (size_t)(m0 + l16) * lda + 2 * half;
  const float* bp = B + (size_t)half * ldb + n0 + l16;
#pragma unroll 4
  for (int k = 0; k < K; k += 4) {
    v2f afrag = { ap[k], ap[k + 1] };
    v2f bfrag = { bp[(size_t)k * ldb], bp[(size_t)(k + 2) * ldb] };
    acc = __builtin_amdgcn_wmma_f32_16x16x4_f32(
        false, afrag, false, bfrag, (short)0, acc, false, false);
  }
  return acc;
}

// ---------------------------------------------------------------------------
// Kernel 1: h[b,n,o] = x[b,n,:] @ W[:,o] + bias[o]
// grid (64, B), block 256 (8 waves); wave -> one 16x16 output tile, K=128.
// ---------------------------------------------------------------------------
__global__ void gat_xw_kernel(const float* __restrict__ x,
                              const float* __restrict__ W,
                              const float* __restrict__ bias,
                              float* __restrict__ h) {
  const int wave = threadIdx.x >> 5;
  const int lane = threadIdx.x & 31;
  const int b    = blockIdx.y;
  const int tile = blockIdx.x * 8 + wave;       // 0..511
  const int mt = tile >> 3, nt = tile & 7;      // 64 M-tiles x 8 N-tiles
  const int m0 = mt * 16, n0 = nt * 16;

  v8f acc = wmma_f32_tile(x + (size_t)b * N_ * IND_, IND_,
                          W, OUTD_, IND_, m0, n0, lane);

  const int half = lane >> 4, l16 = lane & 15;
  const float bv = bias[n0 + l16];
  float* out = h + (size_t)b * N_ * OUTD_;
#pragma unroll
  for (int j = 0; j < 8; ++j)
    out[(size_t)(m0 + j + 8 * half) * OUTD_ + n0 + l16] = acc[j] + bv;
}

// ---------------------------------------------------------------------------
// Kernel 2: si[r] = h[r,:]·a_i, sj[r] = h[r,:]·a_j, one wave32 per row.
// ---------------------------------------------------------------------------
__global__ void gat_sisj_kernel(const float* __restrict__ h,
                                const float* __restrict__ a_i,
                                const float* __restrict__ a_j,
                                float* __restrict__ si,
                                float* __restrict__ sj) {
  const int wave = threadIdx.x >> 5;
  const int lane = threadIdx.x & 31;
  const int row  = blockIdx.x * 8 + wave;       // 0..B*N-1
  const float* hp = h + (size_t)row * OUTD_;
  float di = 0.f, dj = 0.f;
#pragma unroll
  for (int c = lane; c < OUTD_; c += 32) {
    const float v = hp[c];
    di += v * a_i[c];
    dj += v * a_j[c];
  }
#pragma unroll
  for (int off = 16; off > 0; off >>= 1) {
    di += __shfl_down(di, off, 32);
    dj += __shfl_down(dj, off, 32);
  }
  if (lane == 0) { si[row] = di; sj[row] = dj; }
}

// ---------------------------------------------------------------------------
// Kernel 3: fused edge-score + LeakyReLU + adjacency mask + row softmax.
// One 256-thread block per (b,n) row; row of 1024 scores lives in LDS.
// Streams 64KB of edge_attr + 4KB adj per block: this is the HBM floor
// (544MB total @ 23.3TB/s ~= 24us).
// ---------------------------------------------------------------------------
__global__ void gat_attn_kernel(const float* __restrict__ edge_attr,
                                const int*   __restrict__ adj,
                                const float* __restrict__ si,
                                const float* __restrict__ sj,
                                const float* __restrict__ a_e,
                                const float* __restrict__ b_att,
                                float* __restrict__ P) {
  __shared__ float s_e[N_];
  __shared__ float s_red[256];
  __shared__ float s_ae[ED_];

  const int b   = blockIdx.y;
  const int n   = blockIdx.x;
  const int tid = threadIdx.x;
  const size_t row = (size_t)b * N_ + n;

  if (tid < ED_) s_ae[tid] = a_e[tid];
  __syncthreads();

  const float base = si[row] + b_att[0];
  const float* __restrict__ sjb    = sj + (size_t)b * N_;
  const int*   __restrict__ adjrow = adj + row * N_;
  const float* __restrict__ ea     = edge_attr + row * (size_t)N_ * ED_;

  float lmax = -INFINITY;
#pragma unroll
  for (int m = tid; m < N_; m += 256) {
    const float4* ep = (const float4*)(ea + (size_t)m * ED_);
    const float4 e0 = ep[0], e1 = ep[1], e2 = ep[2], e3 = ep[3];
    float se = e0.x * s_ae[0]  + e0.y * s_ae[1]  + e0.z * s_ae[2]  + e0.w * s_ae[3]
             + e1.x * s_ae[4]  + e1.y * s_ae[5]  + e1.z * s_ae[6]  + e1.w * s_ae[7]
             + e2.x * s_ae[8]  + e2.y * s_ae[9]  + e2.z * s_ae[10] + e2.w * s_ae[11]
             + e3.x * s_ae[12] + e3.y * s_ae[13] + e3.z * s_ae[14] + e3.w * s_ae[15];
    float v = base + sjb[m] + se;
    v = (v > 0.f) ? v : (ALPHA_ * v);          // LeakyReLU
    v = (adjrow[m] > 0) ? v : NEG_INF_;        // adjacency mask
    s_e[m] = v;
    lmax = fmaxf(lmax, v);
  }

  s_red[tid] = lmax;
  __syncthreads();
#pragma unroll
  for (int s = 128; s > 0; s >>= 1) {
    if (tid < s) s_red[tid] = fmaxf(s_red[tid], s_red[tid + s]);
    __syncthreads();
  }
  const float rmax = s_red[0];
  __syncthreads();

  float lsum = 0.f;
#pragma unroll
  for (int m = tid; m < N_; m += 256) {
    const float ev = expf(s_e[m] - rmax);
    s_e[m] = ev;
    lsum += ev;
  }
  s_red[tid] = lsum;
  __syncthreads();
#pragma unroll
  for (int s = 128; s > 0; s >>= 1) {
    if (tid < s) s_red[tid] += s_red[tid + s];
    __syncthreads();
  }
  const float inv = 1.0f / s_red[0];

  float* __restrict__ prow = P + row * N_;
#pragma unroll
  for (int m = tid; m < N_; m += 256)
    prow[m] = s_e[m] * inv;
}

// ---------------------------------------------------------------------------
// Kernel 4: out = ELU(P @ h), per batch: [1024x1024]x[1024x128], K=1024.
// Block = 8 waves = one 128-row M-stripe x full N=128.
//   - wave w owns rows [stripe*128 + w*16, +16), all 8 N-tiles (64 acc VGPRs)
//   - h staged in LDS as 64x128 chunks (32KB); read once per stripe from L2
//   - each 8B A-fragment load feeds 8 WMMAs -> P is streamed exactly once
// grid (8, B).
// ---------------------------------------------------------------------------
__global__ void gat_ph_kernel(const float* __restrict__ P,
                              const float* __restrict__ h,
                              float* __restrict__ out) {
  __shared__ float sh[KC_ * OUTD_];             // 32 KB

  const int tid  = threadIdx.x;
  const int wave = tid >> 5;
  const int lane = tid & 31;
  const int b    = blockIdx.y;
  const int m0   = blockIdx.x * 128 + wave * 16;
  const int half = lane >> 4, l16 = lane & 15;

  v8f acc[8];
#pragma unroll
  for (int nt = 0; nt < 8; ++nt)
    acc[nt] = (v8f){0.f, 0.f, 0.f, 0.f, 0.f, 0.f, 0.f, 0.f};

  // per-lane A base: row m0+l16, starting K offset 2*half
  const float* __restrict__ Arow =
      P + (size_t)b * N_ * N_ + (size_t)(m0 + l16) * N_ + 2 * half;
  const float* __restrict__ hb = h + (size_t)b * N_ * OUTD_;

  for (int kc = 0; kc < N_; kc += KC_) {
    // cooperative stage: h[b, kc:kc+KC, 0:128] -> LDS (2048 float4 / 256 thr)
    const float4* __restrict__ src = (const float4*)(hb + (size_t)kc * OUTD_);
    float4* __restrict__ dst = (float4*)sh;
#pragma unroll
    for (int i = 0; i < (KC_ * OUTD_ / 4) / 256; ++i)
      dst[tid + i * 256] = src[tid + i * 256];
    __syncthreads();

#pragma unroll 2
    for (int k = 0; k < KC_; k += 4) {
      const v2f afrag = { Arow[kc + k], Arow[kc + k + 1] };
#pragma unroll
      for (int nt = 0; nt < 8; ++nt) {
        const v2f bfrag = { sh[(k + half) * OUTD_ + nt * 16 + l16],
                            sh[(k + half + 2) * OUTD_ + nt * 16 + l16] };
        acc[nt] = __builtin_amdgcn_wmma_f32_16x16x4_f32(
            false, afrag, false, bfrag, (short)0, acc[nt], false, false);
      }
    }
    __syncthreads();
  }

  float* __restrict__ o = out + (size_t)b * N_ * OUTD_;
#pragma unroll
  for (int nt = 0; nt < 8; ++nt) {
#pragma unroll
    for (int j = 0; j < 8; ++j) {
      const float v = acc[nt][j];
      o[(size_t)(m0 + j + 8 * half) * OUTD_ + nt * 16 + l16] =
          (v > 0.f) ? v : expm1f(v);           // ELU (alpha=1)
    }
  }
}

// ---------------------------------------------------------------------------
extern "C" void kernel_launch(void* const* d_in, const int* in_sizes, int n_in,
                              void* d_out, int out_size, void* d_ws, size_t ws_size,
                              hipStream_t stream) {
  (void)in_sizes; (void)n_in; (void)out_size; (void)ws_size;

  const float* x         = (const float*)d_in[0];
  const int*   adj       = (const int*)  d_in[1];
  const float* edge_attr = (const float*)d_in[2];
  const float* W_lin     = (const float*)d_in[3];
  const float* b_lin     = (const float*)d_in[4];
  const float* a_i       = (const float*)d_in[5];
  const float* a_j       = (const float*)d_in[6];
  const float* a_e       = (const float*)d_in[7];
  const float* b_att     = (const float*)d_in[8];
  float*       out       = (float*)d_out;

  // workspace layout (fp32): h | si | sj | P   (~36.1 MB total)
  float* h  = (float*)d_ws;                       // B*N*OUTD  = 1,048,576
  float* si = h  + (size_t)B_ * N_ * OUTD_;       // B*N       = 8,192
  float* sj = si + (size_t)B_ * N_;               // B*N       = 8,192
  float* P  = sj + (size_t)B_ * N_;               // B*N*N     = 8,388,608

  // 1) h = xW + b
  gat_xw_kernel<<<dim3(64, B_), 256, 0, stream>>>(x, W_lin, b_lin, h);

  // 2) si, sj
  gat_sisj_kernel<<<(B_ * N_) / 8, 256, 0, stream>>>(h, a_i, a_j, si, sj);

  // 3) fused scores + mask + softmax
  gat_attn_kernel<<<dim3(N_, B_), 256, 0, stream>>>(edge_attr, adj, si, sj,
                                                    a_e, b_att, P);

  // 4) out = ELU(P @ h), LDS-blocked
  gat_ph_kernel<<<dim3(8, B_), 256, 0, stream>>>(P, h, out);
}